// QuaRotR4_72713796322123
// MI455X (gfx1250) — compile-verified
//
#include <hip/hip_runtime.h>
#include <hip/hip_bf16.h>

typedef __attribute__((ext_vector_type(2))) float v2f;
typedef __attribute__((ext_vector_type(8))) float v8f;

#define CHUNK   512
#define KROWS   28
#define HIDDEN  (CHUNK * KROWS)   /* 14336 */
#define LDSTR   516               /* padded LDS row stride in floats: 516 % 64 == 4 */

__global__ __launch_bounds__(256)
void quarot_fused_kernel(const float* __restrict__ X,
                         const float* __restrict__ H,
                         float* __restrict__ Y,
                         int n_t) {
    const int t = blockIdx.x;
    if (t >= n_t) return;
    const int lane = threadIdx.x & 31;
    const int wave = threadIdx.x >> 5;

    __shared__ __align__(16) float lds[KROWS * LDSTR];

    const float scale = 1.0f / 119.73303f;   // 1/sqrt(14336)

    // ---------- Stage 1: FWHT-512 per chunk, register-resident ----------
    // element index e = lane*16 + r  (lane bits = high 5 bits, r = low 4 bits)
    for (int j = wave; j < KROWS; j += 8) {
        const float4* src4 = reinterpret_cast<const float4*>(
            X + (size_t)t * HIDDEN + j * CHUNK + lane * 16);
        float r[16];
        #pragma unroll
        for (int q = 0; q < 4; ++q) {
            float4 v = src4[q];
            r[4*q+0] = v.x; r[4*q+1] = v.y; r[4*q+2] = v.z; r[4*q+3] = v.w;
        }
        // strides 1,2,4,8: in-lane butterflies
        #pragma unroll
        for (int h = 1; h < 16; h <<= 1) {
            #pragma unroll
            for (int i = 0; i < 16; i += 2*h) {
                #pragma unroll
                for (int k = 0; k < h; ++k) {
                    float a = r[i+k], b = r[i+k+h];
                    r[i+k]   = a + b;
                    r[i+k+h] = a - b;
                }
            }
        }
        // strides 16,32,64,128,256: cross-lane via shfl_xor (wave32)
        #pragma unroll
        for (int m = 1; m <= 16; m <<= 1) {
            #pragma unroll
            for (int k = 0; k < 16; ++k) {
                float o = __shfl_xor(r[k], m, 32);
                r[k] = (lane & m) ? (o - r[k]) : (r[k] + o);
            }
        }
        #pragma unroll
        for (int k = 0; k < 16; ++k) r[k] *= scale;

        float4* dst4 = reinterpret_cast<float4*>(&lds[j * LDSTR + lane * 16]);
        #pragma unroll
        for (int q = 0; q < 4; ++q) {
            float4 v;
            v.x = r[4*q+0]; v.y = r[4*q+1]; v.z = r[4*q+2]; v.w = r[4*q+3];
            dst4[q] = v;
        }
    }
    __syncthreads();

    // ---------- Stage 2: Y[i,c] = sum_j H[i][j] * Xh[j,c] via v_wmma_f32_16x16x4_f32 ----------
    // A operand (16x4 f32): lanes 0-15 -> {K0,K1}, lanes 16-31 -> {K2,K3}, M = lane&15
    const int am = lane & 15;
    const int kh = (lane >> 4) << 1;      // 0 or 2
    v2f A[2][7];
    #pragma unroll
    for (int mt = 0; mt < 2; ++mt) {
        int mrow = mt * 16 + am;
        #pragma unroll
        for (int kt = 0; kt < 7; ++kt) {
            int kb = kt * 4 + kh;
            float ax = 0.0f, ay = 0.0f;
            if (mrow < KROWS) {           // pad rows 28..31 with zeros
                ax = H[mrow * KROWS + kb];
                ay = H[mrow * KROWS + kb + 1];
            }
            v2f a; a.x = ax; a.y = ay;
            A[mt][kt] = a;
        }
    }

    // B operand (4x16 f32): lanes 0-15 -> {K0,K1}, lanes 16-31 -> {K2,K3}, N = lane&15
    const int ccol  = lane & 15;
    const int cbase = wave * 64;          // each wave owns 4 N-tiles of 16 columns
    const int mro   = (lane < 16) ? 0 : 8;
    float* outt = Y + (size_t)t * HIDDEN;

    for (int nt = 0; nt < 4; ++nt) {
        const int c = cbase + nt * 16 + ccol;
        v8f acc0 = {};   // M rows 0..15
        v8f acc1 = {};   // M rows 16..31 (28..31 discarded)
        #pragma unroll
        for (int kt = 0; kt < 7; ++kt) {
            int j0 = kt * 4 + kh;
            v2f b;
            b.x = lds[j0 * LDSTR + c];
            b.y = lds[(j0 + 1) * LDSTR + c];
            acc0 = __builtin_amdgcn_wmma_f32_16x16x4_f32(
                       false, A[0][kt], false, b, (short)0, acc0, false, false);
            acc1 = __builtin_amdgcn_wmma_f32_16x16x4_f32(
                       false, A[1][kt], false, b, (short)0, acc1, false, false);
        }
        // D layout: VGPR v holds M = v + (lane<16 ? 0 : 8), N = lane&15
        float* o = outt + c;
        #pragma unroll
        for (int v = 0; v < 8; ++v) {
            o[(v + mro) * CHUNK] = acc0[v];
            int m1 = 16 + v + mro;
            if (m1 < KROWS) o[m1 * CHUNK] = acc1[v];
        }
    }
}

extern "C" void kernel_launch(void* const* d_in, const int* in_sizes, int n_in,
                              void* d_out, int out_size, void* d_ws, size_t ws_size,
                              hipStream_t stream) {
    const float* X = (const float*)d_in[0];
    const float* H = (const float*)d_in[1];
    float*       Y = (float*)d_out;
    const int n_t = in_sizes[0] / HIDDEN;   // 4096
    dim3 grid(n_t), block(256);
    hipLaunchKernelGGL(quarot_fused_kernel, grid, block, 0, stream, X, H, Y, n_t);
}